// RejectionSampler_43198781063594
// MI455X (gfx1250) — compile-verified
//
#include <hip/hip_runtime.h>
#include <stdint.h>

#define BS 128
#define K 8
#define VOCAB 50257
#define NDRAFT (BS * K)

#define TILE2 2048                        // floats per tile (256 lanes x 8 floats = 8KB)
#define DEPTH 4                           // async pipeline ring depth (tiles)
#define NT2 ((VOCAB + TILE2 - 1) / TILE2) // 25 tiles per row
#define ROWY_PAD 50260                    // VOCAB rounded up to multiple of 4

#define SEED_U 0x13579BDFu
#define SEED_REC 0x2468ACE1u
#define SEED_BONUS 0x0F1E2D3Cu

// ---------------- deterministic counter-based RNG ----------------
__device__ __forceinline__ uint32_t mix32(uint32_t x) {
  x ^= x >> 16; x *= 0x7FEB352Du;
  x ^= x >> 15; x *= 0x846CA68Bu;
  x ^= x >> 16;
  return x;
}
__device__ __forceinline__ float u01(uint32_t seed, uint32_t a, uint32_t b) {
  uint32_t h = mix32(seed ^ mix32(a * 0x9E3779B1u + mix32(b ^ 0x85EBCA77u)));
  return ((float)(h >> 8) + 0.5f) * (1.0f / 16777216.0f);  // strictly in (0,1)
}

// ---------------- CDNA5 async global->LDS (ASYNCcnt path) ----------------
typedef __attribute__((address_space(3))) float lds_float;

__device__ __forceinline__ void async_copy_f4(const float* gsrc, float* ldst) {
  uint32_t loff = (uint32_t)(uintptr_t)(lds_float*)ldst;     // LDS byte offset
  unsigned long long gaddr = (unsigned long long)(uintptr_t)gsrc;
  asm volatile("global_load_async_to_lds_b128 %0, %1, off"
               :
               : "v"(loff), "v"(gaddr)
               : "memory");
}
template <int N>
__device__ __forceinline__ void wait_async() {
  asm volatile("s_wait_asynccnt %0" ::"n"(N) : "memory");
}
// 2 async ops per tile; allow R tiles to stay in flight, guarantee current tile done.
__device__ __forceinline__ void wait_tiles_inflight(int R) {
  switch (R) {
    case 3: wait_async<6>(); break;
    case 2: wait_async<4>(); break;
    case 1: wait_async<2>(); break;
    default: wait_async<0>(); break;
  }
}

__device__ __forceinline__ int clampe(int e) { return e > (VOCAB - 4) ? (VOCAB - 4) : e; }

// issue one 8KB tile (2 x b128 per lane), tail chunks address-clamped (EXEC-uniform)
__device__ __forceinline__ void issue_tile(const float* row, int t, float* tilebuf, int tid) {
  const int e = t * TILE2 + tid * 8;
  async_copy_f4(row + clampe(e), tilebuf + tid * 8);
  async_copy_f4(row + clampe(e + 4), tilebuf + tid * 8 + 4);
}

// =====================================================================
// Kernel A: bonus token = argmax(logits/temp + Gumbel) over vocab
// =====================================================================
__global__ __launch_bounds__(256) void rs_bonus_kernel(const float* __restrict__ logits,
                                                       const float* __restrict__ temps,
                                                       int* __restrict__ bonus_out) {
  __shared__ float buf[DEPTH][TILE2];
  __shared__ float sk[256];
  __shared__ int si[256];
  const int b = blockIdx.x;
  const int tid = threadIdx.x;
  const float invT = 1.0f / temps[b];
  const float* x = logits + (size_t)(NDRAFT + b) * VOCAB;

  for (int t = 0; t < DEPTH - 1 && t < NT2; ++t) issue_tile(x, t, &buf[t][0], tid);

  float best = -3.4e38f;
  int arg = 0;
  for (int t = 0; t < NT2; ++t) {
    if (t + DEPTH - 1 < NT2) issue_tile(x, t + DEPTH - 1, &buf[(t + DEPTH - 1) & (DEPTH - 1)][0], tid);
    int R = NT2 - 1 - t; if (R > DEPTH - 1) R = DEPTH - 1;
    wait_tiles_inflight(R);
    const float* tb = &buf[t & (DEPTH - 1)][0];
#pragma unroll
    for (int c = 0; c < 2; ++c) {
      const int e = t * TILE2 + tid * 8 + 4 * c;
      const int ec = clampe(e);
      const float* sl = tb + tid * 8 + 4 * c;
#pragma unroll
      for (int j = 0; j < 4; ++j) {
        int v = e + j;
        if (v < VOCAB) {
          float val = sl[v - ec];
          float uu = u01(SEED_BONUS, (uint32_t)b, (uint32_t)v);
          float g = -__logf(-__logf(uu));  // Gumbel(0,1)
          float key = val * invT + g;
          if (key > best) { best = key; arg = v; }
        }
      }
    }
  }
  sk[tid] = best; si[tid] = arg;
  __syncthreads();
  for (int s = 128; s > 0; s >>= 1) {
    if (tid < s && sk[tid + s] > sk[tid]) { sk[tid] = sk[tid + s]; si[tid] = si[tid + s]; }
    __syncthreads();
  }
  if (tid == 0) bonus_out[b] = si[0];
}

// =====================================================================
// Kernel B: per (b,k) draft row. CDNA5 plan: the whole scaled-logit row
// (196KB) persists in the 320KB WGP LDS, so target logits touch HBM once.
//   phase 1 : async-stream x, store y = x/T into rowY, track row max
//   phase 1b: LDS-only sweep -> sum(exp(y-M)) -> logZ
//   phase 2 : async-stream draft_probs, read y from LDS; residual mass,
//             twin Gumbel argmax (shared exponential-race noise), gathers
// =====================================================================
__global__ __launch_bounds__(256) void rs_target_kernel(
    const float* __restrict__ logits, const float* __restrict__ dprobs,
    const float* __restrict__ temps, const int* __restrict__ draft_ids,
    float* __restrict__ pt_out, float* __restrict__ pd_out, int* __restrict__ rec_out) {
  extern __shared__ float smem[];
  float* rowY = smem;                        // ROWY_PAD floats (196.3 KB)
  float* bufS = rowY + ROWY_PAD;             // DEPTH*TILE2 floats (32 KB)
  float* sa = bufS + DEPTH * TILE2;          // 256
  float* sb = sa + 256;                      // 256
  float* sc = sb + 256;                      // 256
  int* si = (int*)(sc + 256);                // 256
  __shared__ float rowM, rowL;

  const int r = blockIdx.x;
  const int b = r / K;
  const int tid = threadIdx.x;
  const float invT = 1.0f / temps[b];
  const int tok = draft_ids[r];
  const float* x = logits + (size_t)r * VOCAB;
  const float* dp = dprobs + (size_t)r * VOCAB;

  // ---------------- phase 1: stream x once; y -> LDS; row max ----------------
  for (int t = 0; t < DEPTH - 1 && t < NT2; ++t) issue_tile(x, t, bufS + (t & (DEPTH - 1)) * TILE2, tid);
  float m = -3.4e38f;
  for (int t = 0; t < NT2; ++t) {
    if (t + DEPTH - 1 < NT2) issue_tile(x, t + DEPTH - 1, bufS + ((t + DEPTH - 1) & (DEPTH - 1)) * TILE2, tid);
    int R = NT2 - 1 - t; if (R > DEPTH - 1) R = DEPTH - 1;
    wait_tiles_inflight(R);
    const float* tb = bufS + (t & (DEPTH - 1)) * TILE2;
#pragma unroll
    for (int c = 0; c < 2; ++c) {
      const int e = t * TILE2 + tid * 8 + 4 * c;
      const int ec = clampe(e);
      const float* sl = tb + tid * 8 + 4 * c;
      float4 y4;  // positions ec..ec+3 are always valid; duplicates store equal values
      y4.x = sl[0] * invT; y4.y = sl[1] * invT; y4.z = sl[2] * invT; y4.w = sl[3] * invT;
      *(float4*)&rowY[ec] = y4;
#pragma unroll
      for (int j = 0; j < 4; ++j) {
        int v = e + j;
        if (v < VOCAB) m = fmaxf(m, sl[v - ec] * invT);
      }
    }
  }
  sa[tid] = m;
  __syncthreads();
  for (int st = 128; st > 0; st >>= 1) {
    if (tid < st) sa[tid] = fmaxf(sa[tid], sa[tid + st]);
    __syncthreads();
  }
  if (tid == 0) rowM = sa[0];
  __syncthreads();
  const float M = rowM;

  // ---------------- phase 1b: LDS-only sum of exp(y - M) ----------------
  float ssum = 0.0f;
  for (int v = tid; v < VOCAB; v += 256) ssum += __expf(rowY[v] - M);
  sa[tid] = ssum;
  __syncthreads();
  for (int st = 128; st > 0; st >>= 1) {
    if (tid < st) sa[tid] += sa[tid + st];
    __syncthreads();
  }
  if (tid == 0) rowL = __logf(sa[0]);
  __syncthreads();
  const float logZ = rowL;

  // ---------------- phase 2: stream draft_probs; y from LDS ----------------
  for (int t = 0; t < DEPTH - 1 && t < NT2; ++t) issue_tile(dp, t, bufS + (t & (DEPTH - 1)) * TILE2, tid);
  float mass = 0.0f;
  float best1 = -3.4e38f; int arg1 = 0;  // residual distribution
  float best2 = -3.4e38f; int arg2 = 0;  // fallback: target distribution (same noise)
  float ptTok = 0.0f, pdTok = 0.0f;

  auto consume = [&](int v, float y, float d) {
    float lp = y - M - logZ;   // exact log target prob
    float p = __expf(lp);
    float rec = p - d;
    rec = rec > 0.0f ? rec : 0.0f;
    mass += rec;
    // Gumbel-max via exponential race: argmax(log w + G) == argmax(w / e), e~Exp(1)
    float uu = u01(SEED_REC, (uint32_t)r, (uint32_t)v);
    float ie = __builtin_amdgcn_rcpf(-__logf(uu));
    float k1 = rec * ie;
    if (k1 > best1) { best1 = k1; arg1 = v; }
    float k2 = p * ie;
    if (k2 > best2) { best2 = k2; arg2 = v; }
    if (v == tok) { ptTok = p; pdTok = d; }
  };

  for (int t = 0; t < NT2; ++t) {
    if (t + DEPTH - 1 < NT2) issue_tile(dp, t + DEPTH - 1, bufS + ((t + DEPTH - 1) & (DEPTH - 1)) * TILE2, tid);
    int R = NT2 - 1 - t; if (R > DEPTH - 1) R = DEPTH - 1;
    wait_tiles_inflight(R);
    const float* tb = bufS + (t & (DEPTH - 1)) * TILE2;
#pragma unroll
    for (int c = 0; c < 2; ++c) {
      const int e = t * TILE2 + tid * 8 + 4 * c;
      const int ec = clampe(e);
      const float* sl = tb + tid * 8 + 4 * c;
      if (e + 4 <= VOCAB) {
        float4 y4 = *(const float4*)&rowY[e];
        consume(e + 0, y4.x, sl[0]);
        consume(e + 1, y4.y, sl[1]);
        consume(e + 2, y4.z, sl[2]);
        consume(e + 3, y4.w, sl[3]);
      } else {
#pragma unroll
        for (int j = 0; j < 4; ++j) {
          int v = e + j;
          if (v < VOCAB) consume(v, rowY[v], sl[v - ec]);
        }
      }
    }
  }

  // sums: residual mass, p_t(tok), p_d(tok)
  sa[tid] = mass; sb[tid] = ptTok; sc[tid] = pdTok;
  __syncthreads();
  for (int st = 128; st > 0; st >>= 1) {
    if (tid < st) { sa[tid] += sa[tid + st]; sb[tid] += sb[tid + st]; sc[tid] += sc[tid + st]; }
    __syncthreads();
  }
  float massTot = 0.0f, ptV = 0.0f, pdV = 0.0f;
  if (tid == 0) { massTot = sa[0]; ptV = sb[0]; pdV = sc[0]; }
  __syncthreads();
  // argmax over residual keys
  sa[tid] = best1; si[tid] = arg1;
  __syncthreads();
  for (int st = 128; st > 0; st >>= 1) {
    if (tid < st && sa[tid + st] > sa[tid]) { sa[tid] = sa[tid + st]; si[tid] = si[tid + st]; }
    __syncthreads();
  }
  int a1 = 0;
  if (tid == 0) a1 = si[0];
  __syncthreads();
  // argmax over target keys (fallback when residual mass == 0)
  sa[tid] = best2; si[tid] = arg2;
  __syncthreads();
  for (int st = 128; st > 0; st >>= 1) {
    if (tid < st && sa[tid + st] > sa[tid]) { sa[tid] = sa[tid + st]; si[tid] = si[tid + st]; }
    __syncthreads();
  }
  if (tid == 0) {
    rec_out[r] = (massTot > 0.0f) ? a1 : si[0];
    pt_out[r] = ptV;
    pd_out[r] = pdV;
  }
}

// =====================================================================
// Kernel C: accept/reject per row, assemble [BS, K+1] output (-1 padded)
// =====================================================================
__global__ __launch_bounds__(128) void rs_finalize_kernel(
    const int* __restrict__ draft_ids, const float* __restrict__ pt,
    const float* __restrict__ pd, const int* __restrict__ rec_tok,
    const int* __restrict__ bonus_tok, int* __restrict__ out) {
  int b = blockIdx.x * blockDim.x + threadIdx.x;
  if (b >= BS) return;
  int nacc = 0;
  bool rejected = false;
#pragma unroll
  for (int k = 0; k < K; ++k) {
    float p_t = pt[b * K + k], p_d = pd[b * K + k];
    float ap = (p_d > 0.0f) ? fminf(1.0f, p_t / fmaxf(p_d, 1e-30f)) : 0.0f;
    float u = u01(SEED_U, (uint32_t)b, (uint32_t)k);
    if (!rejected && u <= ap) nacc++; else rejected = true;
  }
  int rej = (nacc < K - 1) ? nacc : (K - 1);
  int fin = (nacc == K) ? bonus_tok[b] : rec_tok[b * K + rej];
#pragma unroll
  for (int pos = 0; pos <= K; ++pos) {
    int o = (pos < nacc) ? draft_ids[b * K + pos] : ((pos == nacc) ? fin : -1);
    out[b * (K + 1) + pos] = o;
  }
}

extern "C" void kernel_launch(void* const* d_in, const int* in_sizes, int n_in,
                              void* d_out, int out_size, void* d_ws, size_t ws_size,
                              hipStream_t stream) {
  (void)in_sizes; (void)n_in; (void)out_size; (void)ws_size;
  const int* draft_ids = (const int*)d_in[0];
  const float* logits = (const float*)d_in[1];
  const float* temps = (const float*)d_in[2];
  const float* dprobs = (const float*)d_in[3];
  int* out = (int*)d_out;

  // workspace: pt[1024] f32 | pd[1024] f32 | rec_tok[1024] i32 | bonus[128] i32
  float* pt = (float*)d_ws;
  float* pd = pt + NDRAFT;
  int* rtok = (int*)(pd + NDRAFT);
  int* btok = rtok + NDRAFT;

  // dynamic LDS for kernel B: 200KB row + 32KB stream ring + reductions (~232KB of 320KB WGP LDS)
  constexpr size_t RS_SMEM_BYTES =
      sizeof(float) * (ROWY_PAD + DEPTH * TILE2 + 3 * 256) + sizeof(int) * 256;

  rs_bonus_kernel<<<BS, 256, 0, stream>>>(logits, temps, btok);
  rs_target_kernel<<<NDRAFT, 256, RS_SMEM_BYTES, stream>>>(logits, dprobs, temps, draft_ids,
                                                           pt, pd, rtok);
  rs_finalize_kernel<<<1, 128, 0, stream>>>(draft_ids, pt, pd, rtok, btok, out);
}